// GNNRetriever_71760313581919
// MI455X (gfx1250) — compile-verified
//
#include <hip/hip_runtime.h>
#include <hip/hip_bf16.h>

typedef __attribute__((ext_vector_type(2))) float v2f;
typedef __attribute__((ext_vector_type(8))) float v8f;

#define N_NODES 50000
#define N_EDGES 800000
#define DIM 128

// ---------------------------------------------------------------------------
// Degree / dinv kernels
// ---------------------------------------------------------------------------
__global__ void gcn_init_deg(float* __restrict__ deg, int n) {
    int i = blockIdx.x * blockDim.x + threadIdx.x;
    if (i < n) deg[i] = 1.0f;  // self-loop
}

__global__ void gcn_deg_accum(const int* __restrict__ dst, float* __restrict__ deg, int ne) {
    int e = blockIdx.x * blockDim.x + threadIdx.x;
    if (e < ne) atomicAdd(&deg[dst[e]], 1.0f);
}

__global__ void gcn_rsqrt_inplace(float* __restrict__ deg, int n) {
    int i = blockIdx.x * blockDim.x + threadIdx.x;
    if (i < n) deg[i] = rsqrtf(deg[i]);
}

// ---------------------------------------------------------------------------
// WMMA f32 GEMM: H[M,128] = X[M,128] @ W[128,128]
// One wave computes one 16x16 tile of H. K loop: 32 x V_WMMA_F32_16X16X4_F32.
// A 16x4 f32 layout: lane l holds row M=l%16, K = {2*(l/16), 2*(l/16)+1}.
// B 4x16 f32 layout: lane l holds col N=l%16, same K split.
// C/D 16x16 f32:   c[v] is element (M = v + 8*(l/16), N = l%16).
// ---------------------------------------------------------------------------
__global__ void gcn_gemm_wmma(const float* __restrict__ X, const float* __restrict__ W,
                              float* __restrict__ H, int nrows) {
    const int wavesPerBlock = blockDim.x >> 5;
    const int wave = blockIdx.x * wavesPerBlock + (threadIdx.x >> 5);
    const int lane = threadIdx.x & 31;
    const int NT = DIM / 16;              // 8 tiles along N
    const int mt = wave / NT;
    const int nt = wave % NT;
    if (mt * 16 >= nrows) return;         // wave-uniform guard (never taken here)

    const int lo = lane & 15;             // M (for A) or N (for B) within tile
    const int kh = (lane >> 4) << 1;      // 0 or 2: K sub-offset for this half-wave

    const float* __restrict__ xrow = X + (size_t)(mt * 16 + lo) * DIM;
    const float* __restrict__ wcol = W + (size_t)(nt * 16 + lo);

    v8f c = {0.f, 0.f, 0.f, 0.f, 0.f, 0.f, 0.f, 0.f};
#pragma unroll
    for (int k = 0; k < DIM; k += 4) {
        v2f a, b;
        a.x = xrow[k + kh];
        a.y = xrow[k + kh + 1];
        b.x = wcol[(size_t)(k + kh) * DIM];
        b.y = wcol[(size_t)(k + kh + 1) * DIM];
        c = __builtin_amdgcn_wmma_f32_16x16x4_f32(
            /*neg_a=*/false, a, /*neg_b=*/false, b,
            /*c_mod=*/(short)0, c, /*reuse_a=*/false, /*reuse_b=*/false);
    }

    const int mbase = mt * 16 + ((lane >> 4) << 3);   // +8 for upper half-wave
    float* __restrict__ out = H + (size_t)mbase * DIM + nt * 16 + lo;
#pragma unroll
    for (int v = 0; v < 8; ++v) {
        out[(size_t)v * DIM] = c[v];
    }
}

// ---------------------------------------------------------------------------
// Zero a float buffer (float4-wide)
// ---------------------------------------------------------------------------
__global__ void gcn_zero_f4(float4* __restrict__ p, int n4) {
    int i = blockIdx.x * blockDim.x + threadIdx.x;
    if (i < n4) p[i] = make_float4(0.f, 0.f, 0.f, 0.f);
}

// ---------------------------------------------------------------------------
// Edge scatter: one wave per edge; lane owns one float4 (4 channels) of row.
// src/dst/coef are wave-uniform -> scalar loads; data path is 512B/wave.
// ---------------------------------------------------------------------------
__global__ void gcn_edge_agg(const float* __restrict__ H,
                             const int* __restrict__ src, const int* __restrict__ dst,
                             const float* __restrict__ dinv,
                             float* __restrict__ AGG, int ne) {
    const int e = blockIdx.x * (blockDim.x >> 5) + (threadIdx.x >> 5);
    const int lane = threadIdx.x & 31;
    if (e >= ne) return;
    const int s = src[e];
    const int d = dst[e];
    const float coef = dinv[s] * dinv[d];

    const float4* __restrict__ hrow = (const float4*)(H + (size_t)s * DIM);
    float* __restrict__ arow = AGG + (size_t)d * DIM;
    float4 hv = hrow[lane];
    atomicAdd(&arow[lane * 4 + 0], hv.x * coef);
    atomicAdd(&arow[lane * 4 + 1], hv.y * coef);
    atomicAdd(&arow[lane * 4 + 2], hv.z * coef);
    atomicAdd(&arow[lane * 4 + 3], hv.w * coef);
}

// ---------------------------------------------------------------------------
// Epilogue: AGG[i][c] = (AGG[i][c] + H[i][c]*dinv[i]^2 + b[c]) , optional ReLU.
// float4-wide, in-place on AGG.
// ---------------------------------------------------------------------------
__global__ void gcn_finalize(float4* __restrict__ AGG, const float4* __restrict__ H,
                             const float* __restrict__ dinv, const float4* __restrict__ bias,
                             int n, int do_relu) {
    int idx = blockIdx.x * blockDim.x + threadIdx.x;   // over n * (DIM/4)
    int n4 = n * (DIM / 4);
    if (idx >= n4) return;
    int row = idx >> 5;          // DIM/4 == 32 float4 per row
    int c4 = idx & 31;
    float di = dinv[row];
    float di2 = di * di;
    float4 a = AGG[idx];
    float4 h = H[idx];
    float4 b = bias[c4];
    float rx = a.x + h.x * di2 + b.x;
    float ry = a.y + h.y * di2 + b.y;
    float rz = a.z + h.z * di2 + b.z;
    float rw = a.w + h.w * di2 + b.w;
    if (do_relu) {
        rx = fmaxf(rx, 0.f); ry = fmaxf(ry, 0.f);
        rz = fmaxf(rz, 0.f); rw = fmaxf(rw, 0.f);
    }
    AGG[idx] = make_float4(rx, ry, rz, rw);
}

// ---------------------------------------------------------------------------
// Launcher
// ---------------------------------------------------------------------------
extern "C" void kernel_launch(void* const* d_in, const int* in_sizes, int n_in,
                              void* d_out, int out_size, void* d_ws, size_t ws_size,
                              hipStream_t stream) {
    const float* x  = (const float*)d_in[0];
    const int*   ei = (const int*)d_in[1];          // [2, N_EDGES] flat
    const float* W1 = (const float*)d_in[2];
    const float* b1 = (const float*)d_in[3];
    const float* W2 = (const float*)d_in[4];
    const float* b2 = (const float*)d_in[5];
    float* out = (float*)d_out;

    const int* src = ei;
    const int* dst = ei + N_EDGES;

    // Workspace layout (256B-aligned chunks)
    char* ws = (char*)d_ws;
    float* dinv = (float*)ws;                               // 50000 f  (200 KB)
    float* H    = (float*)(ws + 256 * ((N_NODES * 4 + 255) / 256));            // 25.6 MB
    float* AGG  = H + (size_t)N_NODES * DIM;                                    // 25.6 MB

    const int T = 256;
    const int nodeBlocks = (N_NODES + T - 1) / T;
    const int edgeBlocks = (N_EDGES + T - 1) / T;
    const int featN4     = N_NODES * (DIM / 4);             // 1.6M float4
    const int featBlocks = (featN4 + T - 1) / T;
    const int gemmWaves  = (N_NODES / 16) * (DIM / 16);     // 25000 (exact)
    const int gemmBlocks = gemmWaves / 4;                   // 4 waves / 128-thread block
    const int aggBlocks  = (N_EDGES * 32 + T - 1) / T;      // 1 wave per edge

    // --- degrees ---
    gcn_init_deg<<<nodeBlocks, T, 0, stream>>>(dinv, N_NODES);
    gcn_deg_accum<<<edgeBlocks, T, 0, stream>>>(dst, dinv, N_EDGES);
    gcn_rsqrt_inplace<<<nodeBlocks, T, 0, stream>>>(dinv, N_NODES);

    // --- layer 1 ---
    gcn_gemm_wmma<<<gemmBlocks, 128, 0, stream>>>(x, W1, H, N_NODES);
    gcn_zero_f4<<<featBlocks, T, 0, stream>>>((float4*)AGG, featN4);
    gcn_edge_agg<<<aggBlocks, T, 0, stream>>>(H, src, dst, dinv, AGG, N_EDGES);
    gcn_finalize<<<featBlocks, T, 0, stream>>>((float4*)AGG, (const float4*)H, dinv,
                                               (const float4*)b1, N_NODES, /*relu=*/1);

    // --- layer 2 (AGG now holds X2; aggregate directly into d_out) ---
    gcn_gemm_wmma<<<gemmBlocks, 128, 0, stream>>>(AGG, W2, H, N_NODES);
    gcn_zero_f4<<<featBlocks, T, 0, stream>>>((float4*)out, featN4);
    gcn_edge_agg<<<aggBlocks, T, 0, stream>>>(H, src, dst, dinv, out, N_EDGES);
    gcn_finalize<<<featBlocks, T, 0, stream>>>((float4*)out, (const float4*)H, dinv,
                                               (const float4*)b2, N_NODES, /*relu=*/0);
}